// Graph_17540646436884
// MI455X (gfx1250) — compile-verified
//
#include <hip/hip_runtime.h>

#define D 128
#define TILE 16

typedef __attribute__((ext_vector_type(2))) float v2f;
typedef __attribute__((ext_vector_type(8))) float v8f;

// ---------------------------------------------------------------------------
// Guaranteed-native fp32 global atomic add, no return value (STOREcnt path),
// device scope so scatter-adds from all WGPs are coherent at L2.
// ---------------------------------------------------------------------------
__device__ __forceinline__ void atomAddF32(float* p, float v) {
    asm volatile("global_atomic_add_f32 %0, %1, off scope:SCOPE_DEV"
                 :: "v"(p), "v"(v)
                 : "memory");
}

// ---------------------------------------------------------------------------
// Zero the aggregation buffer (ws is poisoned before timing).
// ---------------------------------------------------------------------------
__global__ void zero_kernel(float4* __restrict__ p, int n4) {
    int i = blockIdx.x * blockDim.x + threadIdx.x;
    if (i < n4) {
        float4 z;
        z.x = 0.f; z.y = 0.f; z.z = 0.f; z.w = 0.f;
        p[i] = z;
    }
}

// ---------------------------------------------------------------------------
// Scatter: one wave per edge. Lane l handles channels [4l, 4l+4).
// h[src] row (512B) is read coalesced as float4 per lane; 4 fp32 no-return
// atomics per lane into agg[dst]. Both h and agg (51.2 MB each) fit in the
// 192 MB L2, so this is L2-atomic-throughput bound, not HBM bound.
// ---------------------------------------------------------------------------
__global__ __launch_bounds__(256)
void scatter_kernel(const float* __restrict__ h,
                    const int*   __restrict__ esrc,
                    const int*   __restrict__ edst,
                    const float* __restrict__ ew,
                    float*       __restrict__ agg,
                    int E) {
    int tid  = blockIdx.x * blockDim.x + threadIdx.x;
    int e    = tid >> 5;
    if (e >= E) return;
    int lane = tid & 31;

    int   s = esrc[e];
    int   d = edst[e];
    float w = ew[e];

    const float4* hv = reinterpret_cast<const float4*>(h + (size_t)s * D);
    float4 v = hv[lane];

    float* out = agg + (size_t)d * D + lane * 4;
    atomAddF32(out + 0, v.x * w);
    atomAddF32(out + 1, v.y * w);
    atomAddF32(out + 2, v.z * w);
    atomAddF32(out + 3, v.w * w);
}

// ---------------------------------------------------------------------------
// Fused GEMM epilogue: out = agg @ Wrel^T + h @ Wroot^T + b  (+optional ReLU)
//
// Wave-level WMMA f32 16x16x4:
//   block = 256 threads = 8 waves; block b covers rows [16b, 16b+16),
//   wave w covers cols [16w, 16w+16) -> one block does 16 rows x 128 cols.
//
// A-layout (16x4 f32): lanes 0-15 hold M=lane, K={kb,kb+1};
//                      lanes 16-31 hold M=lane-16, K={kb+2,kb+3}
//   -> per-lane float2 load at (row, kb + 2*(lane>=16)) — always 8B aligned.
// B-layout (4x16): same striping with N=lane; B[k][n] = W[n*D + k] gives W^T.
// C/D-layout: VGPR v -> row = v + 8*(lane>=16), col = lane&15.
//
// N = 100000 is divisible by 16 -> no row guards, EXEC all-ones for WMMA.
// ---------------------------------------------------------------------------
__global__ __launch_bounds__(256, 2)
void gemm_kernel(const float* __restrict__ Agg,
                 const float* __restrict__ H,
                 const float* __restrict__ Wrel,
                 const float* __restrict__ Wroot,
                 const float* __restrict__ bias,
                 float*       __restrict__ Out,
                 int relu) {
    const int wave = threadIdx.x >> 5;
    const int lane = threadIdx.x & 31;
    const int lh   = lane & 15;
    const int koff = (lane >> 4) << 1;   // 0 for lanes 0-15, 2 for lanes 16-31
    const int row0 = blockIdx.x * TILE;
    const int col0 = wave * TILE;

    const float* arow  = Agg   + (size_t)(row0 + lh) * D;
    const float* hrow  = H     + (size_t)(row0 + lh) * D;
    const float* wrelr = Wrel  + (size_t)(col0 + lh) * D;
    const float* wrotr = Wroot + (size_t)(col0 + lh) * D;

    v8f acc = {};

#pragma unroll
    for (int kb = 0; kb < D; kb += 4) {
        v2f a = *reinterpret_cast<const v2f*>(arow  + kb + koff);
        v2f b = *reinterpret_cast<const v2f*>(wrelr + kb + koff);
        acc = __builtin_amdgcn_wmma_f32_16x16x4_f32(
            /*neg_a=*/false, a, /*neg_b=*/false, b,
            /*c_mod=*/(short)0, acc, /*reuse_a=*/false, /*reuse_b=*/false);
    }
#pragma unroll
    for (int kb = 0; kb < D; kb += 4) {
        v2f a = *reinterpret_cast<const v2f*>(hrow  + kb + koff);
        v2f b = *reinterpret_cast<const v2f*>(wrotr + kb + koff);
        acc = __builtin_amdgcn_wmma_f32_16x16x4_f32(
            false, a, false, b, (short)0, acc, false, false);
    }

    const float bv   = bias[col0 + lh];
    const int   moff = (lane >> 4) << 3;   // 0 or 8

#pragma unroll
    for (int v = 0; v < 8; ++v) {
        float o = acc[v] + bv;
        if (relu) o = fmaxf(o, 0.0f);
        Out[(size_t)(row0 + v + moff) * D + col0 + lh] = o;
    }
}

// ---------------------------------------------------------------------------
// Host launcher: 3 layers of {zero agg, scatter, fused WMMA GEMM}.
// ws layout: [ agg | hA | hB ]  (each N*D floats = 51.2 MB)
// ---------------------------------------------------------------------------
extern "C" void kernel_launch(void* const* d_in, const int* in_sizes, int n_in,
                              void* d_out, int out_size, void* d_ws, size_t ws_size,
                              hipStream_t stream) {
    const float* x   = (const float*)d_in[0];
    const float* ew  = (const float*)d_in[1];
    const float* Wrel[3]  = {(const float*)d_in[2], (const float*)d_in[5], (const float*)d_in[8]};
    const float* Wroot[3] = {(const float*)d_in[3], (const float*)d_in[6], (const float*)d_in[9]};
    const float* bias[3]  = {(const float*)d_in[4], (const float*)d_in[7], (const float*)d_in[10]};
    const int* esrc = (const int*)d_in[11];
    const int* edst = (const int*)d_in[12];

    const int N = in_sizes[0] / D;   // 100000
    const int E = in_sizes[1];       // 1600000

    float* agg = (float*)d_ws;
    float* hA  = agg + (size_t)N * D;
    float* hB  = hA  + (size_t)N * D;
    float* out = (float*)d_out;

    const float* hin = x;
    float* houts[3] = {hA, hB, out};

    const int n4 = (N * D) / 4;
    dim3 zblk(256), zgrid((n4 + 255) / 256);
    dim3 sblk(256), sgrid((E + 7) / 8);          // 8 waves/block, 1 edge/wave
    dim3 gblk(256), ggrid((N + TILE - 1) / TILE);

    for (int l = 0; l < 3; ++l) {
        zero_kernel<<<zgrid, zblk, 0, stream>>>((float4*)agg, n4);
        scatter_kernel<<<sgrid, sblk, 0, stream>>>(hin, esrc, edst, ew, agg, E);
        gemm_kernel<<<ggrid, gblk, 0, stream>>>(agg, hin, Wrel[l], Wroot[l],
                                                bias[l], houts[l],
                                                (l < 2) ? 1 : 0);
        hin = houts[l];
    }
}